// SWAPLayer_2379411882436
// MI455X (gfx1250) — compile-verified
//
#include <hip/hip_runtime.h>
#include <cstdint>

// SWAP gate on qubits (0,1) of 13 qubits == permutation: out[i] = in[swap bits 12,11 of i].
// Roofline: permutation copy = 16 MB traffic (~0.7us @ 23.3 TB/s) vs WMMA matmul
// needing >= 256 MB just to read U (>= 11us). So: streaming permutation copy through
// the CDNA5 async global<->LDS data mover (ASYNCcnt path), now with the bounds guards
// removed from the bulk kernel (grid divides the problem exactly) to kill the
// per-op exec-mask branching seen in round 1's disasm.

#define THREADS        256
#define F4_PER_THREAD  4
#define F4_PER_BLOCK   (THREADS * F4_PER_THREAD)   // 1024 float4 per block
// Per batch row: 8192 floats = 2048 float4. Element bits 12,11 == float4-index bits 10,9.

__device__ __forceinline__ unsigned permute_src_f4(unsigned g)
{
    unsigned row = g >> 11;                       // batch row (2048 f4 each)
    unsigned w   = g & 2047u;                     // f4 index within row
    unsigned d   = ((w >> 9) ^ (w >> 10)) & 1u;   // swap f4 bits 9,10
    unsigned ws  = w ^ ((d << 9) | (d << 10));
    return (row << 11) | ws;                      // source f4 index
}

__global__ __launch_bounds__(THREADS)
void swap01_bulk(const float* __restrict__ in, float* __restrict__ out)
{
    __shared__ float4 stage[F4_PER_BLOCK];   // 16 KB: private 16B slot per (lane, k)

    const unsigned t    = threadIdx.x;
    const unsigned base = blockIdx.x * F4_PER_BLOCK;

    unsigned lds_off[F4_PER_THREAD];
    uint64_t dst_addr[F4_PER_THREAD];

    // Phase 1: async global->LDS loads from permuted source addresses (coalesced:
    // consecutive lanes hit consecutive 16B lines; the bit-9/10 swap only reorders
    // 8KB-aligned (512-float4) chunks, never breaks contiguity within a wave).
    #pragma unroll
    for (int k = 0; k < F4_PER_THREAD; ++k) {
        unsigned g = base + t + (unsigned)k * THREADS;
        unsigned s = permute_src_f4(g);

        uint64_t src_addr = (uint64_t)(uintptr_t)in + (uint64_t)s * 16ull;
        dst_addr[k]       = (uint64_t)(uintptr_t)out + (uint64_t)g * 16ull;
        // Low 32 bits of a generic pointer into LDS == LDS byte offset.
        lds_off[k] = (unsigned)(uintptr_t)(&stage[t + (unsigned)k * THREADS]);

        asm volatile("global_load_async_to_lds_b128 %0, %1, off"
                     :: "v"(lds_off[k]), "v"(src_addr)
                     : "memory");
    }

    // ASYNCcnt decrements only when the LDS write has completed -> one wait, no
    // barrier needed (slots are lane-private).
    asm volatile("s_wait_asynccnt 0x0" ::: "memory");

    // Phase 2: async LDS->global stores to contiguous destinations.
    #pragma unroll
    for (int k = 0; k < F4_PER_THREAD; ++k) {
        asm volatile("global_store_async_from_lds_b128 %0, %1, off"
                     :: "v"(dst_addr[k]), "v"(lds_off[k])
                     : "memory");
    }
    // S_ENDPGM's implicit wait-idle drains outstanding async stores.
}

// Tail kernel (guarded, plain VMEM): only launched if total4 is not a multiple of
// F4_PER_BLOCK. For the reference sizes (524288 f4) it never runs.
__global__ __launch_bounds__(THREADS)
void swap01_tail(const float* __restrict__ in, float* __restrict__ out,
                 unsigned start4, unsigned total4)
{
    unsigned g = start4 + blockIdx.x * THREADS + threadIdx.x;
    if (g < total4) {
        const float4* in4  = (const float4*)in;
        float4*       out4 = (float4*)out;
        out4[g] = in4[permute_src_f4(g)];
    }
}

extern "C" void kernel_launch(void* const* d_in, const int* in_sizes, int n_in,
                              void* d_out, int out_size, void* d_ws, size_t ws_size,
                              hipStream_t stream) {
    const float* in  = (const float*)d_in[0];   // [256, 8192, 1] f32
    float*       out = (float*)d_out;           // [256, 8192, 1] f32
    // d_in[1] is U (8192x8192): intentionally unused — it is a known permutation.

    unsigned total4      = (unsigned)(out_size / 4);          // 524288 float4
    unsigned bulk_blocks = total4 / F4_PER_BLOCK;             // 512 (exact)
    unsigned bulk4       = bulk_blocks * F4_PER_BLOCK;

    if (bulk_blocks) {
        swap01_bulk<<<bulk_blocks, THREADS, 0, stream>>>(in, out);
    }
    if (bulk4 < total4) {
        unsigned rem = total4 - bulk4;
        unsigned tb  = (rem + THREADS - 1) / THREADS;
        swap01_tail<<<tb, THREADS, 0, stream>>>(in, out, bulk4, total4);
    }
}